// Net_19628000543029
// MI455X (gfx1250) — compile-verified
//
#include <hip/hip_runtime.h>
#include <hip/hip_bf16.h>
#include <math.h>

// Net: B=512 graphs, P=96 nodes, F=5, H=64, K=16 neighbors, O=10 classes.
#define NB 512
#define NP 96
#define NF 5
#define NH 64
#define NK 16
#define NO 10
#define XS 66   // lx stride (floats): even (8B-aligned pairs), bank = 2*row+k
#define AS 68   // A-staging stride: bank = 4m+k -> half-waves (k vs k+2) disjoint

typedef __attribute__((ext_vector_type(2))) float v2f;
typedef __attribute__((ext_vector_type(8))) float v8f;

__device__ __forceinline__ v8f wmma_f32_k4(v2f a, v2f b, v8f c) {
  // D = A(16x4,f32) * B(4x16,f32) + C(16x16,f32)  -- exact fp32 accumulate
  return __builtin_amdgcn_wmma_f32_16x16x4_f32(false, a, false, b, (short)0, c,
                                               false, false);
}

__device__ __forceinline__ float elu(float v) {
  return v > 0.0f ? v : (__expf(v) - 1.0f);
}

__device__ __forceinline__ void lds_wave_fence() {
  // wave-local LDS RAW: per-wave LDS ops complete in order; wait DScnt==0 and
  // stop the compiler from reordering around it.
  asm volatile("s_wait_dscnt 0x0" ::: "memory");
  __builtin_amdgcn_wave_barrier();
}

// Pre-paired weight layout: lwp[pswz(p,n)] = { W[2p][n], W[2p+1][n] } (v2f).
// A B-fragment (k even) is then ONE aligned ds_load_b64. The XOR on bit4 of n
// by pair parity puts lower-half (pair p) and upper-half (pair p+1) reads in
// disjoint 32-bank halves -> conflict-free.
__device__ __forceinline__ int pswz(int p, int n) {
  return p * NH + (n ^ ((p & 1) << 4));
}

// ---------------------------------------------------------------------------
// Kernel 1: input MLP  x*datanorm -> ELU(5->64) -> ELU(64->64) -> ELU(64->64)
// 128 threads = 4 waves; each wave owns 16 rows; layers 1/2 use f32 WMMA with
// kk-outer / nt-inner (4 interleaved accumulator chains).
// ---------------------------------------------------------------------------
__global__ __launch_bounds__(128) void input_mlp_kernel(
    const float* __restrict__ x, const float* __restrict__ dn,
    const float* __restrict__ W0, const float* __restrict__ b0,
    const float* __restrict__ W1, const float* __restrict__ b1,
    const float* __restrict__ W2, const float* __restrict__ b2,
    float* __restrict__ out) {
  __shared__ v2f lw1p[32 * NH];       // 64x64 pre-paired
  __shared__ v2f lw2p[32 * NH];       // 64x64 pre-paired
  __shared__ float lhA[4][16 * AS];
  __shared__ float lhB[4][16 * AS];

  const int tid = threadIdx.x;
  const int wave = tid >> 5;
  const int lane = tid & 31;
  const int m = lane & 15;
  const int hi = lane >> 4;
  const int koff = hi * 2;

  for (int t = tid; t < 32 * NH; t += 128) {
    const int p = t >> 6, n = t & 63;
    v2f w1; w1.x = W1[(2 * p) * NH + n]; w1.y = W1[(2 * p + 1) * NH + n];
    v2f w2; w2.x = W2[(2 * p) * NH + n]; w2.y = W2[(2 * p + 1) * NH + n];
    lw1p[pswz(p, n)] = w1;
    lw2p[pswz(p, n)] = w2;
  }
  __syncthreads();

  const int rowbase = blockIdx.x * 64 + wave * 16;
  const int row = rowbase + m;

  // ---- layer 0 (K=5, scalar): each half-wave covers 32 of the 64 outputs
  float xv[NF];
#pragma unroll
  for (int f = 0; f < NF; ++f) xv[f] = x[row * NF + f] * dn[f];
#pragma unroll
  for (int c = 0; c < 32; ++c) {
    const int n = hi * 32 + c;
    float acc = b0[n];
#pragma unroll
    for (int f = 0; f < NF; ++f) acc = fmaf(xv[f], W0[f * NH + n], acc);
    lhA[wave][m * AS + n] = elu(acc);
  }
  lds_wave_fence();

  // ---- layer 1: [16x64] @ [64x64], lhA -> lhB
  {
    v8f acc[4];
#pragma unroll
    for (int nt = 0; nt < 4; ++nt) {
      const float bb = b1[nt * 16 + m];
#pragma unroll
      for (int r = 0; r < 8; ++r) acc[nt][r] = bb;
    }
#pragma unroll
    for (int kk = 0; kk < 16; ++kk) {
      const int k = kk * 4 + koff;        // even
      const int p = k >> 1;               // pair index = 2*kk + hi
      const v2f a = *(const v2f*)&lhA[wave][m * AS + k];
#pragma unroll
      for (int nt = 0; nt < 4; ++nt)
        acc[nt] = wmma_f32_k4(a, lw1p[pswz(p, nt * 16 + m)], acc[nt]);
    }
#pragma unroll
    for (int nt = 0; nt < 4; ++nt)
#pragma unroll
      for (int r = 0; r < 8; ++r)
        lhB[wave][(r + hi * 8) * AS + nt * 16 + m] = elu(acc[nt][r]);
  }
  lds_wave_fence();

  // ---- layer 2: [16x64] @ [64x64], lhB -> global
  {
    v8f acc[4];
#pragma unroll
    for (int nt = 0; nt < 4; ++nt) {
      const float bb = b2[nt * 16 + m];
#pragma unroll
      for (int r = 0; r < 8; ++r) acc[nt][r] = bb;
    }
#pragma unroll
    for (int kk = 0; kk < 16; ++kk) {
      const int k = kk * 4 + koff;
      const int p = k >> 1;
      const v2f a = *(const v2f*)&lhB[wave][m * AS + k];
#pragma unroll
      for (int nt = 0; nt < 4; ++nt)
        acc[nt] = wmma_f32_k4(a, lw2p[pswz(p, nt * 16 + m)], acc[nt]);
    }
#pragma unroll
    for (int nt = 0; nt < 4; ++nt)
#pragma unroll
      for (int r = 0; r < 8; ++r)
        out[(rowbase + r + hi * 8) * NH + nt * 16 + m] = elu(acc[nt][r]);
  }
}

// ---------------------------------------------------------------------------
// Kernel 2: kNN (K=16, no self-loops) within each graph of 96 nodes.
// One block (96 threads) per graph; x_i cached in VGPRs; register top-16.
// ---------------------------------------------------------------------------
__global__ __launch_bounds__(96) void knn_kernel(const float* __restrict__ h,
                                                 int* __restrict__ idx) {
  __shared__ float lx[NP * XS];
  const int g = blockIdx.x;
  const int i = threadIdx.x;  // node id 0..95

  for (int t = i; t < NP * NH; t += NP) {
    const int p = t >> 6, d = t & 63;
    lx[p * XS + d] = h[(g * NP + p) * NH + d];
  }
  __syncthreads();

  float xi[NH];
#pragma unroll
  for (int d = 0; d < NH; ++d) xi[d] = lx[i * XS + d];

  float bd[NK];
  int bi[NK];
#pragma unroll
  for (int s = 0; s < NK; ++s) { bd[s] = 3.4e38f; bi[s] = 0; }

  for (int j = 0; j < NP; ++j) {
    if (j == i) continue;  // reference: +1e9 on diagonal => never picked
    float d2 = 0.0f;
#pragma unroll
    for (int d = 0; d < NH; ++d) {
      const float df = xi[d] - lx[j * XS + d];  // broadcast read of row j
      d2 = fmaf(df, df, d2);
    }
    if (d2 < bd[NK - 1]) {  // sorted (ascending) register insert, unrolled
#pragma unroll
      for (int s = NK - 1; s >= 0; --s) {
        if (d2 < bd[s]) {
          if (s == 0 || d2 >= bd[s - 1]) { bd[s] = d2; bi[s] = j; }
          else { bd[s] = bd[s - 1]; bi[s] = bi[s - 1]; }
        }
      }
    }
  }
#pragma unroll
  for (int s = 0; s < NK; ++s) idx[(g * NP + i) * NK + s] = bi[s];
}

// ---------------------------------------------------------------------------
// Kernel 3: EdgeConv. One block (8 waves) per graph; one wave per node-tile:
// the 16 edges of node i form a 16x128 A tile [x_i | x_j - x_i], built on the
// fly from LDS. GEMM1 (16x128 @ 128x64) -> ELU -> LDS -> GEMM2 (16x64 @ 64x64)
// -> ELU -> row-sum (= sum over K neighbors) -> hout[node].
// kk-outer / nt-inner: 4 independent WMMA chains; all fragments are single
// aligned ds_load_b64s (pre-paired weights, v2f A loads).
// ---------------------------------------------------------------------------
__global__ __launch_bounds__(256) void edgeconv_kernel(
    const float* __restrict__ hin, const int* __restrict__ idx,
    const float* __restrict__ W0, const float* __restrict__ b0,   // [128,64]
    const float* __restrict__ W1, const float* __restrict__ b1,   // [64,64]
    float* __restrict__ hout) {
  __shared__ float lx[NP * XS];        // node features
  __shared__ v2f lw0p[64 * NH];        // 128x64 pre-paired
  __shared__ v2f lw1p[32 * NH];        // 64x64  pre-paired
  __shared__ float lh1[8][16 * AS];    // per-wave ELU(h1) staging (A of GEMM2)

  const int g = blockIdx.x;
  const int tid = threadIdx.x;
  const int wave = tid >> 5;
  const int lane = tid & 31;
  const int m = lane & 15;
  const int hi = lane >> 4;
  const int koff = hi * 2;

  for (int t = tid; t < NP * NH; t += 256) {
    const int p = t >> 6, d = t & 63;
    lx[p * XS + d] = hin[(g * NP + p) * NH + d];
  }
  for (int t = tid; t < 64 * NH; t += 256) {
    const int p = t >> 6, n = t & 63;
    v2f w; w.x = W0[(2 * p) * NH + n]; w.y = W0[(2 * p + 1) * NH + n];
    lw0p[pswz(p, n)] = w;
  }
  for (int t = tid; t < 32 * NH; t += 256) {
    const int p = t >> 6, n = t & 63;
    v2f w; w.x = W1[(2 * p) * NH + n]; w.y = W1[(2 * p + 1) * NH + n];
    lw1p[pswz(p, n)] = w;
  }
  __syncthreads();

  for (int it = 0; it < NP / 8; ++it) {
    const int node = it * 8 + wave;
    const int jm = idx[(g * NP + node) * NK + m];  // neighbor of edge-row m

    // ---- GEMM1: 16x128 @ 128x64, bias + ELU -> lh1[wave]
    {
      v8f acc[4];
#pragma unroll
      for (int nt = 0; nt < 4; ++nt) {
        const float bb = b0[nt * 16 + m];
#pragma unroll
        for (int r = 0; r < 8; ++r) acc[nt][r] = bb;
      }
#pragma unroll
      for (int kk = 0; kk < 32; ++kk) {
        const int k = kk * 4 + koff;      // even
        const int p = k >> 1;             // pair index
        v2f a;
        if (k < NH) {  // cols 0..63: x_i (identical for all 16 edge rows)
          a = *(const v2f*)&lx[node * XS + k];
        } else {       // cols 64..127: x_j - x_i (row gather)
          const v2f xj = *(const v2f*)&lx[jm * XS + (k - NH)];
          const v2f xc = *(const v2f*)&lx[node * XS + (k - NH)];
          a = xj - xc;
        }
#pragma unroll
        for (int nt = 0; nt < 4; ++nt)
          acc[nt] = wmma_f32_k4(a, lw0p[pswz(p, nt * 16 + m)], acc[nt]);
      }
#pragma unroll
      for (int nt = 0; nt < 4; ++nt)
#pragma unroll
        for (int r = 0; r < 8; ++r)
          lh1[wave][(r + hi * 8) * AS + nt * 16 + m] = elu(acc[nt][r]);
    }
    lds_wave_fence();

    // ---- GEMM2: 16x64 @ 64x64, bias + ELU, then sum over the 16 edge rows
    {
      v8f acc[4];
#pragma unroll
      for (int nt = 0; nt < 4; ++nt) {
        const float bb = b1[nt * 16 + m];
#pragma unroll
        for (int r = 0; r < 8; ++r) acc[nt][r] = bb;
      }
#pragma unroll
      for (int kk = 0; kk < 16; ++kk) {
        const int k = kk * 4 + koff;
        const int p = k >> 1;
        const v2f a = *(const v2f*)&lh1[wave][m * AS + k];
#pragma unroll
        for (int nt = 0; nt < 4; ++nt)
          acc[nt] = wmma_f32_k4(a, lw1p[pswz(p, nt * 16 + m)], acc[nt]);
      }
#pragma unroll
      for (int nt = 0; nt < 4; ++nt) {
        float s = 0.0f;
#pragma unroll
        for (int r = 0; r < 8; ++r) s += elu(acc[nt][r]);  // rows hi*8..hi*8+7
        s += __shfl_xor(s, 16, 32);                        // + other half rows
        if (hi == 0) hout[(g * NP + node) * NH + nt * 16 + m] = s;
      }
    }
    lds_wave_fence();  // lh1[wave] is overwritten next iteration
  }
}

// ---------------------------------------------------------------------------
// Kernel 4: global max-pool over P, output MLP (64->64->64->10), log-softmax.
// One block (64 threads) per graph. Tiny tail: negligible cost.
// ---------------------------------------------------------------------------
__global__ __launch_bounds__(64) void output_kernel(
    const float* __restrict__ h,
    const float* __restrict__ W0, const float* __restrict__ b0,
    const float* __restrict__ W1, const float* __restrict__ b1,
    const float* __restrict__ W2, const float* __restrict__ b2,
    float* __restrict__ out) {
  __shared__ float g0[NH], g1[NH], g2[NH], lg[NO];
  const int g = blockIdx.x;
  const int d = threadIdx.x;

  float mx = -3.4e38f;
  for (int p = 0; p < NP; ++p)
    mx = fmaxf(mx, h[(g * NP + p) * NH + d]);
  g0[d] = mx;
  __syncthreads();

  float acc = b0[d];
  for (int k = 0; k < NH; ++k) acc = fmaf(g0[k], W0[k * NH + d], acc);
  g1[d] = elu(acc);
  __syncthreads();

  acc = b1[d];
  for (int k = 0; k < NH; ++k) acc = fmaf(g1[k], W1[k * NH + d], acc);
  g2[d] = elu(acc);
  __syncthreads();

  if (d < NO) {
    acc = b2[d];
    for (int k = 0; k < NH; ++k) acc = fmaf(g2[k], W2[k * NO + d], acc);
    lg[d] = acc;
  }
  __syncthreads();

  if (d < NO) {
    float m2 = -3.4e38f;
    for (int o = 0; o < NO; ++o) m2 = fmaxf(m2, lg[o]);
    float s = 0.0f;
    for (int o = 0; o < NO; ++o) s += __expf(lg[o] - m2);
    out[g * NO + d] = lg[d] - m2 - __logf(s);
  }
}

// ---------------------------------------------------------------------------
extern "C" void kernel_launch(void* const* d_in, const int* in_sizes, int n_in,
                              void* d_out, int out_size, void* d_ws,
                              size_t ws_size, hipStream_t stream) {
  const float* x      = (const float*)d_in[0];
  const float* dn     = (const float*)d_in[1];
  const float* W_in0  = (const float*)d_in[2];
  const float* b_in0  = (const float*)d_in[3];
  const float* W_in1  = (const float*)d_in[4];
  const float* b_in1  = (const float*)d_in[5];
  const float* W_in2  = (const float*)d_in[6];
  const float* b_in2  = (const float*)d_in[7];
  const float* W_mp00 = (const float*)d_in[8];
  const float* b_mp00 = (const float*)d_in[9];
  const float* W_mp01 = (const float*)d_in[10];
  const float* b_mp01 = (const float*)d_in[11];
  const float* W_mp10 = (const float*)d_in[12];
  const float* b_mp10 = (const float*)d_in[13];
  const float* W_mp11 = (const float*)d_in[14];
  const float* b_mp11 = (const float*)d_in[15];
  const float* W_out0 = (const float*)d_in[16];
  const float* b_out0 = (const float*)d_in[17];
  const float* W_out1 = (const float*)d_in[18];
  const float* b_out1 = (const float*)d_in[19];
  const float* W_out2 = (const float*)d_in[20];
  const float* b_out2 = (const float*)d_in[21];

  // workspace: h_a, h_b (B*P*H f32 each) + idx (B*P*K i32) ~= 28.3 MB
  float* h_a = (float*)d_ws;
  float* h_b = h_a + (size_t)NB * NP * NH;
  int*   ixb = (int*)(h_b + (size_t)NB * NP * NH);

  input_mlp_kernel<<<(NB * NP) / 64, 128, 0, stream>>>(
      x, dn, W_in0, b_in0, W_in1, b_in1, W_in2, b_in2, h_a);

  knn_kernel<<<NB, NP, 0, stream>>>(h_a, ixb);
  edgeconv_kernel<<<NB, 256, 0, stream>>>(h_a, ixb, W_mp00, b_mp00, W_mp01,
                                          b_mp01, h_b);

  knn_kernel<<<NB, NP, 0, stream>>>(h_b, ixb);
  edgeconv_kernel<<<NB, 256, 0, stream>>>(h_b, ixb, W_mp10, b_mp10, W_mp11,
                                          b_mp11, h_a);

  output_kernel<<<NB, NH, 0, stream>>>(h_a, W_out0, b_out0, W_out1, b_out1,
                                       W_out2, b_out2, (float*)d_out);
}